// GNN_35046933135754
// MI455X (gfx1250) — compile-verified
//
#include <hip/hip_runtime.h>
#include <hip/hip_bf16.h>
#include <math.h>

#define D 64
#define NGRAPHS 64
#define NUM_LAYERS 3
#define LSTR 68   // padded LDS row stride (floats): bank = row*4 + col -> conflict-free column reads

typedef float v2f __attribute__((ext_vector_type(2)));
typedef float v8f __attribute__((ext_vector_type(8)));

__device__ __forceinline__ v8f wmma4(v2f a, v2f b, v8f c) {
  // D = A(16x4, f32) * B(4x16, f32) + C(16x16, f32)
  return __builtin_amdgcn_wmma_f32_16x16x4_f32(false, a, false, b, (short)0, c, false, false);
}

// monotonic float<->unsigned mapping so unsigned atomicMax == float max
__device__ __forceinline__ unsigned f2mono(float f) {
  unsigned b = __float_as_uint(f);
  return (b & 0x80000000u) ? ~b : (b | 0x80000000u);
}
__device__ __forceinline__ float mono2f(unsigned m) {
  unsigned b = (m & 0x80000000u) ? (m ^ 0x80000000u) : ~m;
  return __uint_as_float(b);
}
#define AGG_SENTINEL 0x007FFFFFu   // f2mono(-inf)

__device__ __forceinline__ float mish_f(float v) {
  float sp = (v > 20.f) ? v : log1pf(__expf(v));
  return v * tanhf(sp);
}

// ---------------- scatter-max aggregation ----------------

__global__ void init_agg_kernel(unsigned* __restrict__ uagg, long long n) {
  long long t = (long long)blockIdx.x * blockDim.x + threadIdx.x;
  if (t < n) uagg[t] = AGG_SENTINEL;
}

__global__ void scatter_max_kernel(const float* __restrict__ x,
                                   const int* __restrict__ ei,   // [2, E]
                                   unsigned* __restrict__ uagg,
                                   long long E) {
  long long t = (long long)blockIdx.x * blockDim.x + threadIdx.x;
  if (t >= E * D) return;
  int d = (int)(t & (D - 1));
  long long e = t >> 6;
  int s  = ei[e];
  int dn = ei[E + e];
  float v = x[(long long)s * D + d];
  atomicMax(&uagg[(long long)dn * D + d], f2mono(v));
}

// ---------------- fused GraphConv + update-MLP layer (WMMA fp32) ----------------

__device__ __forceinline__ void load_w_lds(float* __restrict__ dst,
                                           const float* __restrict__ src, int tid) {
  for (int i = tid; i < D * D; i += 128)
    dst[(i >> 6) * LSTR + (i & 63)] = src[i];
}

// acc[4] covers the 16x64 output tile (four 16x16 WMMA tiles along features).
// in: LDS 16xD tile (row = node), w: LDS DxD (row = output feature, i.e. W itself, B = W^T)
__device__ __forceinline__ void gemm_acc(const float* __restrict__ in,
                                         const float* __restrict__ w,
                                         int lane, v8f acc[4]) {
  int n16  = lane & 15;
  int koff = (lane >> 4) << 1;   // lanes 0-15: K+0/K+1, lanes 16-31: K+2/K+3
#pragma unroll
  for (int ks = 0; ks < 16; ++ks) {
    int k = ks * 4 + koff;
    v2f a;
    a.x = in[n16 * LSTR + k];
    a.y = in[n16 * LSTR + k + 1];
#pragma unroll
    for (int ot = 0; ot < 4; ++ot) {
      v2f b;
      b.x = w[(ot * 16 + n16) * LSTR + k];
      b.y = w[(ot * 16 + n16) * LSTR + k + 1];
      acc[ot] = wmma4(a, b, acc[ot]);
    }
  }
}

__device__ __forceinline__ void init_bias(v8f acc[4], const float* __restrict__ bias, int lane) {
  int n16 = lane & 15;
#pragma unroll
  for (int ot = 0; ot < 4; ++ot) {
    float bv = bias[ot * 16 + n16];
#pragma unroll
    for (int r = 0; r < 8; ++r) acc[ot][r] = bv;
  }
}

__device__ __forceinline__ void store_acc(float* __restrict__ out, int lane, const v8f acc[4]) {
  int n16 = lane & 15;
  int mo  = ((lane >> 4) & 1) * 8;
#pragma unroll
  for (int ot = 0; ot < 4; ++ot)
#pragma unroll
    for (int r = 0; r < 8; ++r)
      out[(mo + r) * LSTR + ot * 16 + n16] = acc[ot][r];
}

__global__ __launch_bounds__(128) void layer_kernel(
    float* __restrict__ x, const unsigned* __restrict__ uagg,
    const float* __restrict__ Wrel, const float* __restrict__ brel,
    const float* __restrict__ Wroot,
    const float* __restrict__ W1, const float* __restrict__ b1,
    const float* __restrict__ W2, const float* __restrict__ b2,
    int N) {
  __shared__ float sWa[D * LSTR];          // current weight matrix
  __shared__ float sBias[3 * D];           // brel, b1, b2
  __shared__ float sAct[4][16 * LSTR];     // per-wave x tile (residual source / root input)
  __shared__ float sH[4][16 * LSTR];       // per-wave agg -> h -> t tile

  int tid  = threadIdx.x;
  int wave = tid >> 5;
  int lane = tid & 31;

  int base = (blockIdx.x * 4 + wave) * 16;

  // stage biases + first weight + tiles
  if (tid < D) {
    sBias[tid]           = brel[tid];
    sBias[D + tid]       = b1[tid];
    sBias[2 * D + tid]   = b2[tid];
  }
  load_w_lds(sWa, Wrel, tid);
  for (int i = lane; i < 16 * D; i += 32) {
    int r = i >> 6, c = i & 63;
    int grow = base + r; if (grow > N - 1) grow = N - 1;
    size_t gidx = (size_t)grow * D + c;
    sAct[wave][r * LSTR + c] = x[gidx];
    float f = mono2f(uagg[gidx]);
    if ((__float_as_uint(f) & 0x7f800000u) == 0x7f800000u) f = 0.f;  // empty segment -> 0
    sH[wave][r * LSTR + c] = f;
  }
  __syncthreads();

  // h = agg @ Wrel^T + brel + x @ Wroot^T
  v8f acc[4];
  init_bias(acc, &sBias[0], lane);
  gemm_acc(sH[wave], sWa, lane, acc);
  __syncthreads();
  load_w_lds(sWa, Wroot, tid);
  __syncthreads();
  gemm_acc(sAct[wave], sWa, lane, acc);
  store_acc(sH[wave], lane, acc);
  __syncthreads();

  // t = mish(h @ W1^T + b1)
  load_w_lds(sWa, W1, tid);
  __syncthreads();
  init_bias(acc, &sBias[D], lane);
  gemm_acc(sH[wave], sWa, lane, acc);
#pragma unroll
  for (int ot = 0; ot < 4; ++ot)
#pragma unroll
    for (int r = 0; r < 8; ++r) acc[ot][r] = mish_f(acc[ot][r]);
  store_acc(sH[wave], lane, acc);
  __syncthreads();

  // out = x + t @ W2^T + b2
  load_w_lds(sWa, W2, tid);
  __syncthreads();
  init_bias(acc, &sBias[2 * D], lane);
  gemm_acc(sH[wave], sWa, lane, acc);

  int n16 = lane & 15;
  int mo  = ((lane >> 4) & 1) * 8;
#pragma unroll
  for (int ot = 0; ot < 4; ++ot)
#pragma unroll
    for (int r = 0; r < 8; ++r) {
      int row  = mo + r;
      int grow = base + row;
      float v  = acc[ot][r] + sAct[wave][row * LSTR + ot * 16 + n16];
      if (grow < N) x[(size_t)grow * D + ot * 16 + n16] = v;
    }
}

// ---------------- pooling + readout ----------------

__global__ void zero_kernel(float* __restrict__ p, int n) {
  int t = blockIdx.x * blockDim.x + threadIdx.x;
  if (t < n) p[t] = 0.f;
}

__global__ void pool_kernel(const float* __restrict__ x, const int* __restrict__ batch,
                            float* __restrict__ pooled, int N) {
  long long t = (long long)blockIdx.x * blockDim.x + threadIdx.x;
  if (t >= (long long)N * D) return;
  int d = (int)(t & (D - 1));
  int n = (int)(t >> 6);
  atomicAdd(&pooled[(size_t)batch[n] * D + d], x[t]);
}

__global__ void readout_kernel(const float* __restrict__ pooled,
                               const float* __restrict__ Wr1, const float* __restrict__ br1,
                               const float* __restrict__ Wr2, const float* __restrict__ br2,
                               float* __restrict__ out) {
  int g = threadIdx.x;            // one thread per graph (64)
  float acc = 0.f;
  for (int dd = 0; dd < D; ++dd) {
    float h = br1[dd];
    for (int k = 0; k < D; ++k) h += pooled[g * D + k] * Wr1[dd * D + k];
    acc += mish_f(h) * Wr2[dd];
  }
  out[g] = acc + br2[0];
}

// ---------------- launcher ----------------

extern "C" void kernel_launch(void* const* d_in, const int* in_sizes, int n_in,
                              void* d_out, int out_size, void* d_ws, size_t ws_size,
                              hipStream_t stream) {
  const float* x_in  = (const float*)d_in[0];
  const int*   ei    = (const int*)d_in[1];
  const int*   batch = (const int*)d_in[2];
  const float* Wrel  = (const float*)d_in[3];
  const float* brel  = (const float*)d_in[4];
  const float* Wroot = (const float*)d_in[5];
  const float* W1    = (const float*)d_in[6];
  const float* b1    = (const float*)d_in[7];
  const float* W2    = (const float*)d_in[8];
  const float* b2    = (const float*)d_in[9];
  const float* Wr1   = (const float*)d_in[10];
  const float* br1   = (const float*)d_in[11];
  const float* Wr2   = (const float*)d_in[12];
  const float* br2   = (const float*)d_in[13];

  int N       = in_sizes[0] / D;
  long long E = in_sizes[1] / 2;

  char* ws = (char*)d_ws;
  float*    x      = (float*)ws;                                // N*D f32
  unsigned* uagg   = (unsigned*)(ws + (size_t)N * D * 4);       // N*D u32
  float*    pooled = (float*)(ws + (size_t)N * D * 8);          // NGRAPHS*D f32

  hipMemcpyAsync(x, x_in, (size_t)N * D * sizeof(float), hipMemcpyDeviceToDevice, stream);

  long long nd = (long long)N * D;
  long long ed = E * D;
  const int blk = 256;
  dim3 bInit((unsigned)((nd + blk - 1) / blk));
  dim3 bScat((unsigned)((ed + blk - 1) / blk));
  dim3 bLayer((unsigned)((N + 63) / 64));

  for (int layer = 0; layer < NUM_LAYERS; ++layer) {
    init_agg_kernel<<<bInit, blk, 0, stream>>>(uagg, nd);
    scatter_max_kernel<<<bScat, blk, 0, stream>>>(x, ei, uagg, E);
    layer_kernel<<<bLayer, 128, 0, stream>>>(x, uagg, Wrel, brel, Wroot,
                                             W1, b1, W2, b2, N);
  }

  zero_kernel<<<(NGRAPHS * D + blk - 1) / blk, blk, 0, stream>>>(pooled, NGRAPHS * D);
  pool_kernel<<<bInit, blk, 0, stream>>>(x, batch, pooled, N);
  readout_kernel<<<1, NGRAPHS, 0, stream>>>(pooled, Wr1, br1, Wr2, br2, (float*)d_out);
}